// FGNN_71150428225623
// MI455X (gfx1250) — compile-verified
//
#include <hip/hip_runtime.h>
#include <hip/hip_bf16.h>

// ---------------------------------------------------------------------------
// FGNN fused pipeline for gfx1250 (MI455X), wave32 + V_WMMA_F32_16X16X4_F32,
// with Tensor Data Mover (TDM) staging of the A tile into LDS.
//
// Algebraic restructuring: segment_sum(xu @ W1) == segment_sum(xu) @ W1, so
// ring means are computed from a cheap feature-space segment sum + a tiny
// [363,128]@[128,128] GEMM, and the big pass is ONE fused WMMA kernel doing
// the 8 real GEMMs (mm1 & mm2 complex matmuls) + epilogue.
// ---------------------------------------------------------------------------

typedef __attribute__((ext_vector_type(2))) float        v2f;
typedef __attribute__((ext_vector_type(8))) float        v8f;
typedef __attribute__((ext_vector_type(4))) unsigned int v4u;
typedef __attribute__((ext_vector_type(4))) int          v4i;
typedef __attribute__((ext_vector_type(8))) int          v8i;

#define WH      512
#define NPIX    (WH * WH)           // 262144
#define FIN     128
#define NRINGS  363
#define NRPAD   384
#define SUMSZ   (NRPAD * FIN)       // 49152 floats per re/im sum table

// ws float layout
#define OFF_SUMRE 0
#define OFF_SUMIM (OFF_SUMRE + SUMSZ)
#define OFF_CNT   (OFF_SUMIM + SUMSZ)          // 384 floats
#define OFF_SMRE  (OFF_CNT + NRPAD)
#define OFF_SMIM  (OFF_SMRE + SUMSZ)
#define OFF_WT    (OFF_SMIM + SUMSZ)           // 4 * 128*128 transposed weights
#define ZERO_N    (OFF_CNT + NRPAD)            // zero sums + counts each call

__device__ __forceinline__ int ring_of(int i, int j) {
    float fx = (float)(i - 256);
    float fy = (float)(j - 256);
    return (int)floorf(sqrtf(fx * fx + fy * fy));
}

// Issue a 1-D TDM copy of `nelem` floats from global to LDS (tracked by
// TENSORcnt). D# layout per CDNA5 ISA ch.8: group0 = {count, lds_addr,
// global_addr[56:0], type=2}; group1 = {data_size=4B, tensor_dim0,
// tile_dim0, tensor_dim0_stride}. 6-arg builtin form (clang-23 toolchain).
__device__ __forceinline__ void tdm_load_to_lds(const float* gsrc,
                                                unsigned lds_byte_off,
                                                unsigned nelem) {
    unsigned long long ga = (unsigned long long)(size_t)gsrc;
    v4u g0;
    g0.x = 1u;                                         // count=1 (valid), user mode
    g0.y = lds_byte_off;                               // lds_addr (bytes)
    g0.z = (unsigned)(ga & 0xffffffffull);             // global_addr[31:0]
    g0.w = (unsigned)((ga >> 32) & 0x01ffffffull)      // global_addr[56:32]
         | 0x80000000u;                                // type=2 (bits 127:126)
    v8i g1;
    g1[0] = (int)(2u << 16);                           // data_size=2 -> 4 bytes
    g1[1] = (int)((nelem & 0xffffu) << 16);            // tensor_dim0[15:0]
    g1[2] = (int)((nelem >> 16) & 0xffffu);            // tensor_dim0[31:16]
    g1[3] = (int)((nelem & 0xffffu) << 16);            // tile_dim0 (bits 127:112)
    g1[4] = 0;                                         // tile_dim1/2 = 0 (1-D)
    g1[5] = (int)nelem;                                // tensor_dim0_stride lo
    g1[6] = 0;
    g1[7] = 0;
    v4i zero4 = {};
    v8i zero8 = {};
    __builtin_amdgcn_tensor_load_to_lds(g0, g1, zero4, zero4, zero8, 0);
}

// ---------------- kernel 1: zero the accumulator region ----------------
__global__ void fgnn_zero(float* __restrict__ ws, int n) {
    int idx = blockIdx.x * blockDim.x + threadIdx.x;
    if (idx < n) ws[idx] = 0.0f;
}

// ---------------- kernel 2: transpose 4 weight matrices ----------------
// wT[m][o*128+k] = w[m][k*128+o], m in {w1re, w1im, w2re, w2im}
__global__ void fgnn_transpose(const float* __restrict__ w1re,
                               const float* __restrict__ w1im,
                               const float* __restrict__ w2re,
                               const float* __restrict__ w2im,
                               float* __restrict__ wT) {
    int id = blockIdx.x * blockDim.x + threadIdx.x;   // 4*16384 threads
    int m   = id >> 14;
    int rem = id & 16383;
    int k = rem >> 7;
    int o = rem & 127;
    const float* src = (m == 0) ? w1re : (m == 1) ? w1im : (m == 2) ? w2re : w2im;
    wT[m * 16384 + o * 128 + k] = src[k * 128 + o];
}

// ---------------- kernel 3: hierarchical ring reduction ----------------
// One block = 32x32 pixel tile; tile spans <= 46 rings -> 48 LDS bins.
#define RBINS 48
__global__ __launch_bounds__(256) void fgnn_ring_reduce(
        const float* __restrict__ xre, const float* __restrict__ xim,
        const int* __restrict__ mask,
        float* __restrict__ sum_re, float* __restrict__ sum_im,
        float* __restrict__ cnt) {
    __shared__ float s_re[RBINS * FIN];
    __shared__ float s_im[RBINS * FIN];
    __shared__ float s_ct[RBINS];

    int tid = threadIdx.x;
    int tile_i = (blockIdx.x >> 4) * 32;
    int tile_j = (blockIdx.x & 15) * 32;

    for (int idx = tid; idx < RBINS * FIN; idx += 256) { s_re[idx] = 0.f; s_im[idx] = 0.f; }
    if (tid < RBINS) s_ct[tid] = 0.f;
    __syncthreads();

    // analytic minimum ring in this tile (closest point of the coord box)
    int lox = tile_i - 256, hix = tile_i + 31 - 256;
    int loy = tile_j - 256, hiy = tile_j + 31 - 256;
    float ax = (lox > 0) ? (float)lox : ((hix < 0) ? (float)(-hix) : 0.f);
    float ay = (loy > 0) ? (float)loy : ((hiy < 0) ? (float)(-hiy) : 0.f);
    int rmin = (int)floorf(sqrtf(ax * ax + ay * ay));

    int f   = tid & 127;
    int sub = tid >> 7;    // two pixel-parallel groups of 128 threads
    for (int pp = sub; pp < 1024; pp += 2) {
        int pi = pp >> 5, pj = pp & 31;
        int i = tile_i + pi, j = tile_j + pj;
        int pixel = i * WH + j;
        int bin = ring_of(i, j) - rmin;
        bin = (bin < 0) ? 0 : ((bin > RBINS - 1) ? RBINS - 1 : bin);
        if (f == 0) atomicAdd(&s_ct[bin], 1.0f);
        if (mask[pixel] == 0) {
            atomicAdd(&s_re[bin * FIN + f], xre[(size_t)pixel * FIN + f]);
            atomicAdd(&s_im[bin * FIN + f], xim[(size_t)pixel * FIN + f]);
        }
    }
    __syncthreads();

    for (int idx = tid; idx < RBINS * FIN; idx += 256) {
        int b = idx >> 7;
        int ring = rmin + b;
        if (ring < NRPAD) {
            float vr = s_re[idx], vi = s_im[idx];
            if (vr != 0.f) atomicAdd(&sum_re[ring * FIN + (idx & 127)], vr);
            if (vi != 0.f) atomicAdd(&sum_im[ring * FIN + (idx & 127)], vi);
        }
    }
    if (tid < RBINS) {
        int ring = rmin + tid;
        if (ring < NRPAD && s_ct[tid] != 0.f) atomicAdd(&cnt[ring], s_ct[tid]);
    }
}

// ---------------- kernel 4: tiny GEMM -> per-ring complex means --------
// smean = (Rre @ w1re - Rim @ w1im, Rre @ w1im + Rim @ w1re) / count
__global__ void fgnn_ring_mean(const float* __restrict__ sum_re,
                               const float* __restrict__ sum_im,
                               const float* __restrict__ cnt,
                               const float* __restrict__ wT,
                               float* __restrict__ sm_re,
                               float* __restrict__ sm_im) {
    int id = blockIdx.x * blockDim.x + threadIdx.x;
    if (id >= NRINGS * FIN) return;
    int r = id >> 7, o = id & 127;
    const float* Rre = sum_re + r * FIN;
    const float* Rim = sum_im + r * FIN;
    const float* w1rT = wT + o * 128;            // w1re^T row o
    const float* w1iT = wT + 16384 + o * 128;    // w1im^T row o
    float arr = 0.f, aii = 0.f, ari = 0.f, air = 0.f;
    #pragma unroll 4
    for (int k = 0; k < FIN; ++k) {
        float rr = Rre[k], ri = Rim[k], br = w1rT[k], bi = w1iT[k];
        arr = fmaf(rr, br, arr);
        aii = fmaf(ri, bi, aii);
        ari = fmaf(rr, bi, ari);
        air = fmaf(ri, br, air);
    }
    float d = fmaxf(cnt[r], 1.0f);
    sm_re[r * FIN + o] = (arr - aii) / d;
    sm_im[r * FIN + o] = (ari + air) / d;
}

// ---------------- kernel 5: fused WMMA GEMM + epilogue -----------------
// Block = 256 threads = 8 waves; pixel tile = 16 pixels; wave w owns output
// columns [16w, 16w+16). 8 f32 WMMA accumulators per wave (W1/W2 x rr/ii/ri/ir).
// A tile is staged to LDS via the Tensor Data Mover (TENSORcnt).
__global__ __launch_bounds__(256) void fgnn_fused_wmma(
        const float* __restrict__ xre, const float* __restrict__ xim,
        const int* __restrict__ mask,
        const float* __restrict__ sm_re, const float* __restrict__ sm_im,
        const float* __restrict__ wT,
        float* __restrict__ out) {
    __shared__ float s_xre[16 * FIN];
    __shared__ float s_xim[16 * FIN];
    __shared__ float s_sre[16 * FIN];
    __shared__ float s_sim[16 * FIN];
    __shared__ float s_u[16];
    __shared__ float s_c[16];
    __shared__ int   s_ring[16];

    int tid  = threadIdx.x;
    int lane = tid & 31;
    int wave = tid >> 5;                    // 0..7 = output column tile
    size_t pixelBase = (size_t)blockIdx.x * 16;

    // stage A tile (16 pixels x 128 features, re+im) via TDM async DMA
    if (wave == 0) {
        unsigned lds_re = (unsigned)(size_t)(void*)&s_xre[0];   // low 32b = LDS offset
        unsigned lds_im = (unsigned)(size_t)(void*)&s_xim[0];
        tdm_load_to_lds(xre + pixelBase * FIN, lds_re, 16 * FIN);
        tdm_load_to_lds(xim + pixelBase * FIN, lds_im, 16 * FIN);
    }
    if (tid < 16) {
        size_t p = pixelBase + tid;
        int i = (int)(p >> 9), j = (int)(p & 511);
        int m = mask[p];
        float u = (m == 0) ? 1.0f : 0.0f;
        s_u[tid] = u;
        s_c[tid] = 1.0f - u;                 // mask values are {0,1}
        s_ring[tid] = ring_of(i, j);
    }
    // prefetch next pixel tile while we compute (global_prefetch_b8)
    if (blockIdx.x + 1 < gridDim.x && tid < 64) {
        __builtin_prefetch(xre + (pixelBase + 16) * FIN + tid * 32, 0, 0);
        __builtin_prefetch(xim + (pixelBase + 16) * FIN + tid * 32, 0, 0);
    }
    if (wave == 0) {
        __builtin_amdgcn_s_wait_tensorcnt(0);   // TDM writes to LDS complete
    }
    __syncthreads();
    // gather per-pixel ring means into LDS
    for (int idx = tid; idx < 16 * FIN; idx += 256) {
        int r = s_ring[idx >> 7];
        int f = idx & 127;
        s_sre[idx] = sm_re[r * FIN + f];
        s_sim[idx] = sm_im[r * FIN + f];
    }
    __syncthreads();

    // WMMA fragment mapping (16x16x4 f32):
    //   A: M = lane&15, K pair base = 2*(lane>>4)
    //   B: N = lane&15, same K pair
    int half = lane >> 4;
    int l16  = lane & 15;
    int k2   = 2 * half;
    const float* wt1r = wT;
    const float* wt1i = wT + 16384;
    const float* wt2r = wT + 32768;
    const float* wt2i = wT + 49152;
    int ocol = wave * 16 + l16;             // output feature this lane supplies B for
    float ua = s_u[l16];                    // A-fragment pixel mask weights
    float ca = s_c[l16];

    v8f rr1 = {}, ii1 = {}, ri1 = {}, ir1 = {};
    v8f rr2 = {}, ii2 = {}, ri2 = {}, ir2 = {};

    for (int kk = 0; kk < FIN; kk += 4) {
        int k0 = kk + k2;
        v2f ar = *(const v2f*)&s_xre[l16 * FIN + k0];
        v2f ai = *(const v2f*)&s_xim[l16 * FIN + k0];
        v2f a1r = { ar.x * ua, ar.y * ua };
        v2f a1i = { ai.x * ua, ai.y * ua };
        v2f a2r = { ar.x * ca, ar.y * ca };
        v2f a2i = { ai.x * ca, ai.y * ca };
        v2f b1r = *(const v2f*)&wt1r[ocol * 128 + k0];
        v2f b1i = *(const v2f*)&wt1i[ocol * 128 + k0];
        v2f b2r = *(const v2f*)&wt2r[ocol * 128 + k0];
        v2f b2i = *(const v2f*)&wt2i[ocol * 128 + k0];

        rr1 = __builtin_amdgcn_wmma_f32_16x16x4_f32(false, a1r, false, b1r, (short)0, rr1, false, false);
        ii1 = __builtin_amdgcn_wmma_f32_16x16x4_f32(false, a1i, false, b1i, (short)0, ii1, false, false);
        ri1 = __builtin_amdgcn_wmma_f32_16x16x4_f32(false, a1r, false, b1i, (short)0, ri1, false, false);
        ir1 = __builtin_amdgcn_wmma_f32_16x16x4_f32(false, a1i, false, b1r, (short)0, ir1, false, false);
        rr2 = __builtin_amdgcn_wmma_f32_16x16x4_f32(false, a2r, false, b2r, (short)0, rr2, false, false);
        ii2 = __builtin_amdgcn_wmma_f32_16x16x4_f32(false, a2i, false, b2i, (short)0, ii2, false, false);
        ri2 = __builtin_amdgcn_wmma_f32_16x16x4_f32(false, a2r, false, b2i, (short)0, ri2, false, false);
        ir2 = __builtin_amdgcn_wmma_f32_16x16x4_f32(false, a2i, false, b2r, (short)0, ir2, false, false);
    }

    // epilogue: C/D layout -> row M = v + 8*half, col N = lane&15
    #pragma unroll
    for (int v = 0; v < 8; ++v) {
        int pix = v + 8 * half;
        int f   = wave * 16 + l16;
        float mm1r = rr1[v] - ii1[v];
        float mm1i = ri1[v] + ir1[v];
        float mm2r = rr2[v] - ii2[v];
        float mm2i = ri2[v] + ir2[v];
        float uu = s_u[pix], cc = s_c[pix];
        float sr = s_sre[pix * FIN + f];
        float si = s_sim[pix * FIN + f];
        float outr = uu * (0.5f * (mm1r + sr)) + cc * (mm2r - sr);
        float outi = uu * (0.5f * (mm1i + si)) + cc * (mm2i - si);
        size_t gp = pixelBase + (size_t)pix;
        v2f o2 = { outr, outi };
        *(v2f*)&out[(gp * FIN + f) * 2] = o2;   // [H,W,F,2] {re,im} pair
    }
}

// ---------------------------------------------------------------------------
extern "C" void kernel_launch(void* const* d_in, const int* in_sizes, int n_in,
                              void* d_out, int out_size, void* d_ws, size_t ws_size,
                              hipStream_t stream) {
    const float* xre  = (const float*)d_in[0];
    const float* xim  = (const float*)d_in[1];
    const int*   mask = (const int*)d_in[2];
    // d_in[3], d_in[4] (output_re/im zero buffers) unused
    const float* w1re = (const float*)d_in[5];
    const float* w1im = (const float*)d_in[6];
    const float* w2re = (const float*)d_in[7];
    const float* w2im = (const float*)d_in[8];
    float* out = (float*)d_out;

    float* ws     = (float*)d_ws;     // needs ~1.03 MB: sums, counts, means, wT
    float* sum_re = ws + OFF_SUMRE;
    float* sum_im = ws + OFF_SUMIM;
    float* cnt    = ws + OFF_CNT;
    float* sm_re  = ws + OFF_SMRE;
    float* sm_im  = ws + OFF_SMIM;
    float* wT     = ws + OFF_WT;

    fgnn_zero<<<(ZERO_N + 255) / 256, 256, 0, stream>>>(ws, ZERO_N);
    fgnn_transpose<<<(4 * 16384) / 256, 256, 0, stream>>>(w1re, w1im, w2re, w2im, wT);
    fgnn_ring_reduce<<<(WH / 32) * (WH / 32), 256, 0, stream>>>(xre, xim, mask,
                                                               sum_re, sum_im, cnt);
    fgnn_ring_mean<<<(NRINGS * FIN + 255) / 256, 256, 0, stream>>>(sum_re, sum_im, cnt,
                                                                   wT, sm_re, sm_im);
    fgnn_fused_wmma<<<NPIX / 16, 256, 0, stream>>>(xre, xim, mask, sm_re, sm_im, wT, out);
}